// AdditiveAttention_3899830304894
// MI455X (gfx1250) — compile-verified
//
#include <hip/hip_runtime.h>

// ---------------------------------------------------------------------------
// Additive attention for MI455X (gfx1250, wave32).
//   B=8, NQ=256, NK=512, DQ=DK=DV=256, H=64, fp32 throughout.
//   GEMMs -> V_WMMA_F32_16X16X4_F32 with LDS-staged B (async-to-LDS when
//   available); score/tanh/softmax -> VALU (v_exp/v_rcp).
// ---------------------------------------------------------------------------

typedef __attribute__((ext_vector_type(2))) float v2f;
typedef __attribute__((ext_vector_type(4))) float v4f;
typedef __attribute__((ext_vector_type(8))) float v8f;
typedef __attribute__((ext_vector_type(4))) int   v4i;

#define B_   8
#define NQ_  256
#define NK_  512
#define D_   256
#define H_   64
#define NEG_ (-1e6f)

// LDS B-tile: 32 K-rows x 64 cols, padded to 68 to kill bank conflicts
// between the lane<16 (K,K+1) and lane>=16 (K+2,K+3) halves.
#define BK_      32
#define LDS_PAD_ 68

#if defined(__has_builtin)
#if __has_builtin(__builtin_amdgcn_global_load_async_to_lds_b128) && \
    __has_builtin(__builtin_amdgcn_s_wait_asynccnt)
#define USE_ASYNC_LDS 1
#endif
#endif

// Probe-derived signature: (v4i addrspace(1)*, v4i addrspace(3)*, int, int)
#define GPTR_(p) ((__attribute__((address_space(1))) v4i*)(p))
#define LPTR_(p) ((__attribute__((address_space(3))) v4i*)(p))

__device__ __forceinline__ float fast_tanh(float x) {
  // tanh(x) = 1 - 2/(e^{2x}+1); saturates correctly at +/-1.
  float e = __expf(2.0f * x);
#if __has_builtin(__builtin_amdgcn_rcpf)
  return 1.0f - 2.0f * __builtin_amdgcn_rcpf(e + 1.0f);
#else
  return 1.0f - 2.0f / (e + 1.0f);
#endif
}

// ---------------------------------------------------------------------------
// Band GEMM: a 256-thread block (8 waves) computes a 128 x 64 tile of
// C = A * B.  Wave w owns rows [16w, 16w+16); the 64-wide B strip is staged
// through LDS in 32-deep K chunks and shared by all 8 waves.  Each wave's
// A fragment is reused across the 4 N-subtiles (4 WMMAs per K-step).
//   Abase: this wave's A rows      (LDA row stride, K columns)
//   Bbase: block-uniform B strip   (LDB row stride, 64 columns used)
//   Cbase: this wave's C rows      (LDC row stride, 64 columns)
// ---------------------------------------------------------------------------
template <int K, int LDA, int LDB, int LDC>
__device__ __forceinline__ void gemm_band_64(const float* __restrict__ Abase,
                                             const float* __restrict__ Bbase,
                                             float* __restrict__ Cbase,
                                             float* lds) {
  const int tid  = threadIdx.x;   // 0..255
  const int lane = tid & 31;
  const int half = lane >> 4;     // 0: K{0,1}, 1: K{2,3}
  const int l16  = lane & 15;

  const float* __restrict__ arow = Abase + (size_t)l16 * LDA;
  __builtin_prefetch(arow, 0, 3);              // global_prefetch_b8

  // Staging assignment: thread t loads rows (t>>4) and (t>>4)+16 of the
  // 32 x 64 B chunk, 4 consecutive floats each (one b128 per row).
  const int srow = tid >> 4;           // 0..15
  const int scol = (tid & 15) << 2;    // 0,4,...,60

  v8f acc[4] = {{}, {}, {}, {}};

  for (int kc = 0; kc < K; kc += BK_) {
    const float* g0 = Bbase + (size_t)(kc + srow) * LDB + scol;
    const float* g1 = g0 + (size_t)16 * LDB;
    float* l0 = lds + srow * LDS_PAD_ + scol;
    float* l1 = lds + (srow + 16) * LDS_PAD_ + scol;
#if defined(USE_ASYNC_LDS)
    __builtin_amdgcn_global_load_async_to_lds_b128(GPTR_(g0), LPTR_(l0), 0, 0);
    __builtin_amdgcn_global_load_async_to_lds_b128(GPTR_(g1), LPTR_(l1), 0, 0);
    __builtin_amdgcn_s_wait_asynccnt(0);
#else
    *(v4f*)l0 = *(const v4f*)g0;
    *(v4f*)l1 = *(const v4f*)g1;
#endif
    __syncthreads();

#pragma unroll
    for (int k0 = 0; k0 < BK_; k0 += 4) {
      const int ka = kc + k0 + (half << 1);
      v2f a;
      a.x = arow[ka];
      a.y = arow[ka + 1];
      const float* brow = lds + (k0 + (half << 1)) * LDS_PAD_ + l16;
#pragma unroll
      for (int s = 0; s < 4; ++s) {
        v2f b;
        b.x = brow[s * 16];             // K row (k0 / k0+2)
        b.y = brow[s * 16 + LDS_PAD_];  // K row (k0+1 / k0+3)
        acc[s] = __builtin_amdgcn_wmma_f32_16x16x4_f32(false, a, false, b,
                                                       (short)0, acc[s],
                                                       false, false);
      }
    }
    __syncthreads();   // protect LDS before next chunk overwrites it
  }

#pragma unroll
  for (int s = 0; s < 4; ++s)
#pragma unroll
    for (int r = 0; r < 8; ++r)
      Cbase[(size_t)(r + (half << 3)) * LDC + s * 16 + l16] = acc[s][r];
}

// ---------------------------------------------------------------------------
// Kernel 1: projection  out[M x 64] = A[M x 256] * W[256 x 64].
// One block per 128-row band; grid.x = M/128.
// ---------------------------------------------------------------------------
__global__ void __launch_bounds__(256)
proj_wmma(const float* __restrict__ A, const float* __restrict__ W,
          float* __restrict__ out) {
  __shared__ __align__(16) float tile[BK_ * LDS_PAD_];
  const int wave = threadIdx.x >> 5;
  const int tm = blockIdx.x * 128 + wave * 16;
  gemm_band_64<D_, D_, H_, H_>(A + (size_t)tm * D_, W,
                               out + (size_t)tm * H_, tile);
}

// ---------------------------------------------------------------------------
// Kernel 2: additive scores + masked softmax.
// One block (512 threads) per (b, q) row; thread k owns key k.
//   s[k] = sum_h wv[h] * tanh(qp[b,q,h] + kp[b,k,h]); masked softmax over k.
// ---------------------------------------------------------------------------
__global__ void __launch_bounds__(512)
scores_softmax(const float* __restrict__ qp, const float* __restrict__ kp,
               const float* __restrict__ wv, const int* __restrict__ valid_lens,
               float* __restrict__ attn) {
  const int bq = blockIdx.x;       // 0 .. B*NQ-1
  const int b  = bq >> 8;
  const int k  = threadIdx.x;      // 0 .. 511

  __shared__ float qs[H_];
  __shared__ float ws[H_];
  __shared__ float red[NK_];

  if (threadIdx.x < H_) {
    qs[threadIdx.x] = qp[(size_t)bq * H_ + threadIdx.x];
    ws[threadIdx.x] = wv[threadIdx.x];
  }
  __syncthreads();

  const v4f* __restrict__ krow4 =
      (const v4f*)(kp + ((size_t)b * NK_ + k) * H_);   // 256B-aligned rows
  float s = 0.0f;
#pragma unroll
  for (int i = 0; i < H_ / 4; ++i) {
    const v4f kv = krow4[i];
    s = fmaf(ws[4 * i + 0], fast_tanh(qs[4 * i + 0] + kv.x), s);
    s = fmaf(ws[4 * i + 1], fast_tanh(qs[4 * i + 1] + kv.y), s);
    s = fmaf(ws[4 * i + 2], fast_tanh(qs[4 * i + 2] + kv.z), s);
    s = fmaf(ws[4 * i + 3], fast_tanh(qs[4 * i + 3] + kv.w), s);
  }

  const int valid = valid_lens[b];
  s = (k < valid) ? s : NEG_;

  // block max
  red[k] = s;
  __syncthreads();
#pragma unroll
  for (int off = NK_ / 2; off > 0; off >>= 1) {
    if (k < off) red[k] = fmaxf(red[k], red[k + off]);
    __syncthreads();
  }
  const float m = red[0];
  __syncthreads();

  const float e = __expf(s - m);
  red[k] = e;
  __syncthreads();
#pragma unroll
  for (int off = NK_ / 2; off > 0; off >>= 1) {
    if (k < off) red[k] += red[k + off];
    __syncthreads();
  }
  const float denom = red[0];

#if __has_builtin(__builtin_amdgcn_rcpf)
  attn[(size_t)bq * NK_ + k] = e * __builtin_amdgcn_rcpf(denom);
#else
  attn[(size_t)bq * NK_ + k] = e / denom;
#endif
}

// ---------------------------------------------------------------------------
// Kernel 3: out[b] (256x256) = attn[b] (256x512) * V[b] (512x256).
// Block covers a 128 x 64 tile: grid = B * 2 bands * 4 strips = 64 blocks.
// ---------------------------------------------------------------------------
__global__ void __launch_bounds__(256)
av_wmma(const float* __restrict__ attn, const float* __restrict__ V,
        float* __restrict__ out) {
  __shared__ __align__(16) float tile[BK_ * LDS_PAD_];
  const int b     = blockIdx.x >> 3;
  const int band  = (blockIdx.x >> 2) & 1;
  const int strip = blockIdx.x & 3;
  const int wave  = threadIdx.x >> 5;
  const int tm = band * 128 + wave * 16;
  const int tn = strip * 64;

  const float* Ab = attn + (size_t)b * NQ_ * NK_ + (size_t)tm * NK_;
  const float* Bb = V    + (size_t)b * NK_ * D_ + tn;
  float*       Cb = out  + (size_t)b * NQ_ * D_ + (size_t)tm * D_ + tn;
  gemm_band_64<NK_, NK_, D_, D_>(Ab, Bb, Cb, tile);
}

// ---------------------------------------------------------------------------
// Host-side launch. Inputs (setup_inputs order):
//   0: queries [8,256,256] f32      4: Wq [256,64] f32
//   1: keys    [8,512,256] f32      5: Wk [256,64] f32
//   2: values  [8,512,256] f32      6: wv [64]     f32
//   3: valid_lens [8] i32
// Output: [8,256,256] f32.
// Workspace: qp (128K f) | kp (256K f) | attn (1M f)  ~= 5.8 MB.
// ---------------------------------------------------------------------------
extern "C" void kernel_launch(void* const* d_in, const int* in_sizes, int n_in,
                              void* d_out, int out_size, void* d_ws, size_t ws_size,
                              hipStream_t stream) {
  const float* queries    = (const float*)d_in[0];
  const float* keys       = (const float*)d_in[1];
  const float* values     = (const float*)d_in[2];
  const int*   valid_lens = (const int*)d_in[3];
  const float* Wq         = (const float*)d_in[4];
  const float* Wk         = (const float*)d_in[5];
  const float* wv         = (const float*)d_in[6];
  float* out = (float*)d_out;

  float* qp   = (float*)d_ws;                       // B*NQ*H   = 131072
  float* kp   = qp + (size_t)B_ * NQ_ * H_;         // B*NK*H   = 262144
  float* attn = kp + (size_t)B_ * NK_ * H_;         // B*NQ*NK  = 1048576

  // Q projection: M = B*NQ = 2048 -> 16 bands.
  proj_wmma<<<16, 256, 0, stream>>>(queries, Wq, qp);
  // K projection: M = B*NK = 4096 -> 32 bands.
  proj_wmma<<<32, 256, 0, stream>>>(keys, Wk, kp);
  // Scores + masked softmax: one block per (b,q) row.
  scores_softmax<<<B_ * NQ_, NK_, 0, stream>>>(qp, kp, wv, valid_lens, attn);
  // attn @ V: 64 blocks of 8 waves.
  av_wmma<<<64, 256, 0, stream>>>(attn, values, out);
}